// VectorQuantizer_33079838114250
// MI455X (gfx1250) — compile-verified
//
#include <hip/hip_runtime.h>
#include <stdint.h>

typedef __attribute__((ext_vector_type(2))) float v2f;
typedef __attribute__((ext_vector_type(8))) float v8f;

#define NUM_CODES 8192
#define DIM 128
#define T_LEN 2048
#define B_BATCH 16
#define N_ROWS (B_BATCH * T_LEN)        // 32768
#define CHUNK_CODES 32                  // codes staged in LDS per iteration
#define CHUNK_BYTES (CHUNK_CODES * DIM * 4)   // 16 KB

// ---------------------------------------------------------------------------
// Kernel 1: L2-normalize the codebook rows into workspace (one wave per row).
// ---------------------------------------------------------------------------
__global__ __launch_bounds__(256) void vq_normalize_cb(
    const float* __restrict__ cb, float* __restrict__ cbn) {
  const int lane = threadIdx.x & 31;
  const int row  = blockIdx.x * 8 + (threadIdx.x >> 5);
  const float4* src = (const float4*)(cb + row * DIM);
  float4 v = src[lane];                                   // 4 floats per lane
  float ss = v.x * v.x + v.y * v.y + v.z * v.z + v.w * v.w;
  #pragma unroll
  for (int off = 16; off; off >>= 1) ss += __shfl_xor(ss, off, 32);
  const float rn = 1.0f / fmaxf(sqrtf(ss), 1e-12f);
  float4 o = make_float4(v.x * rn, v.y * rn, v.z * rn, v.w * rn);
  ((float4*)(cbn + row * DIM))[lane] = o;
}

// ---------------------------------------------------------------------------
// Kernel 2: fused cosine-matmul + argmax via V_WMMA_F32_16X16X4_F32.
// Each wave owns 16 consecutive rows (same batch b, consecutive t).
// A-fragment (16x4 f32): lane = r + 16*h holds x[r][4s+2h], x[r][4s+2h+1].
// B-fragment (4x16 f32): lane = c + 16*h holds e[c][4s+2h], e[c][4s+2h+1].
// C (16x16 f32): vgpr j, lane = c + 16*g  ->  dot(row j+8g, col c).
// Two independent accumulator chains (column halves) share the A fragments.
// The codebook streams through a double-buffered LDS ping-pong filled by
// GLOBAL_LOAD_ASYNC_TO_LDS_B128 (ASYNCcnt), so chunk i+1 is fetched by the
// async engine while the 64 WMMAs of chunk i run — no staging VGPRs, no spills.
// ---------------------------------------------------------------------------
__global__ __launch_bounds__(256) void vq_argmax(
    const float* __restrict__ x,        // [B, D, T]
    const float* __restrict__ cbn,      // normalized codebook [K, D]
    int*   __restrict__ idx_i,          // workspace indices [N]
    float* __restrict__ idx_f) {        // output indices [N] (float)
  __shared__ float lds[2][CHUNK_CODES * DIM];   // 2 x 16 KB ping-pong
  const int tid  = threadIdx.x;
  const int wave = tid >> 5;
  const int lane = tid & 31;
  const int r    = lane & 15;           // row / column within tile
  const int h    = lane >> 4;           // half selector
  const int n0   = blockIdx.x * 128 + wave * 16;
  const int b    = n0 >> 11;            // n0 / T_LEN
  const int t0   = n0 & (T_LEN - 1);

  // Per-thread async staging addresses: thread tid covers bytes
  // [tid*16 + i*4096] of each 16 KB chunk (offset: is added to BOTH sides).
  const unsigned lds_base0 = (unsigned)(uintptr_t)(&lds[0][0]) + tid * 16;
  const unsigned lds_base1 = (unsigned)(uintptr_t)(&lds[1][0]) + tid * 16;
  const uint64_t gbase     = (uint64_t)(uintptr_t)cbn + (unsigned)(tid * 16);

  // Prime the pipeline: async-fetch chunk 0 into buffer 0 (4 x b128/thread).
  asm volatile(
      "global_load_async_to_lds_b128 %0, %1, off\n\t"
      "global_load_async_to_lds_b128 %0, %1, off offset:4096\n\t"
      "global_load_async_to_lds_b128 %0, %1, off offset:8192\n\t"
      "global_load_async_to_lds_b128 %0, %1, off offset:12288"
      :: "v"(lds_base0), "v"(gbase) : "memory");

  // Load + L2-normalize this wave's 16-row A tile (32 K-steps of 4);
  // these VMEM loads overlap the first async chunk fetch.
  const float* xb = x + (size_t)b * DIM * T_LEN + t0 + r;
  v2f afrag[32];
  float ss = 0.0f;
  #pragma unroll
  for (int s = 0; s < 32; ++s) {
    const int d = 4 * s + 2 * h;
    float a0 = xb[(size_t)d * T_LEN];
    float a1 = xb[(size_t)(d + 1) * T_LEN];
    afrag[s].x = a0; afrag[s].y = a1;
    ss += a0 * a0 + a1 * a1;
  }
  ss += __shfl_xor(ss, 16, 32);                           // row sum over halves
  const float rn = 1.0f / fmaxf(sqrtf(ss), 1e-12f);
  #pragma unroll
  for (int s = 0; s < 32; ++s) { afrag[s].x *= rn; afrag[s].y *= rn; }

  float best[8];
  int   bidx[8];
  #pragma unroll
  for (int j = 0; j < 8; ++j) { best[j] = -3.402823466e38f; bidx[j] = 0; }

  const int nch = NUM_CODES / CHUNK_CODES;
  for (int ch = 0; ch < nch; ++ch) {
    // Issue the next chunk's async loads into the other buffer. Safe: the
    // end-of-compute barrier of iteration ch-1 guarantees every wave is done
    // reading that buffer. Then retire the 4 loads of THIS chunk
    // (async loads complete in order, so asynccnt<=4 frees the old ones).
    if (ch + 1 < nch) {
      const unsigned d = ((ch + 1) & 1) ? lds_base1 : lds_base0;
      const uint64_t g = gbase + (uint64_t)(ch + 1) * CHUNK_BYTES;
      asm volatile(
          "global_load_async_to_lds_b128 %0, %1, off\n\t"
          "global_load_async_to_lds_b128 %0, %1, off offset:4096\n\t"
          "global_load_async_to_lds_b128 %0, %1, off offset:8192\n\t"
          "global_load_async_to_lds_b128 %0, %1, off offset:12288"
          :: "v"(d), "v"(g) : "memory");
      asm volatile("s_wait_asynccnt 0x4" ::: "memory");
    } else {
      asm volatile("s_wait_asynccnt 0x0" ::: "memory");
    }
    __syncthreads();                                       // chunk ch visible

    const float* buf = lds[ch & 1];
    const float* bb0 = buf + (size_t)r * DIM + 2 * h;          // codes kc..kc+15
    const float* bb1 = buf + (size_t)(16 + r) * DIM + 2 * h;   // codes kc+16..+31
    v8f acc0 = {};
    v8f acc1 = {};
    #pragma unroll
    for (int s = 0; s < 32; ++s) {
      v2f bf0 = *(const v2f*)(bb0 + 4 * s);                    // ds_load_b64
      v2f bf1 = *(const v2f*)(bb1 + 4 * s);
      acc0 = __builtin_amdgcn_wmma_f32_16x16x4_f32(
          false, afrag[s], false, bf0, (short)0, acc0, false, false);
      acc1 = __builtin_amdgcn_wmma_f32_16x16x4_f32(
          false, afrag[s], false, bf1, (short)0, acc1, false, false);
    }
    const int kc   = ch * CHUNK_CODES;
    const int col0 = kc + r;
    const int col1 = kc + 16 + r;
    #pragma unroll
    for (int j = 0; j < 8; ++j) {
      // Merge the two column halves first (tie -> smaller index), then
      // update the single running tracker (strict > keeps earliest code).
      float v; int c;
      if (acc1[j] > acc0[j]) { v = acc1[j]; c = col1; }
      else                   { v = acc0[j]; c = col0; }
      if (v > best[j]) { best[j] = v; bidx[j] = c; }
    }
    __syncthreads();                     // everyone done reading buffer ch&1
  }

  // Reduce across the 16 lanes of each half-group; tie -> smallest index
  // (matches argmin-first semantics of the reference).
  #pragma unroll
  for (int j = 0; j < 8; ++j) {
    #pragma unroll
    for (int off = 8; off; off >>= 1) {
      float ov = __shfl_xor(best[j], off, 32);
      int   oi = __shfl_xor(bidx[j], off, 32);
      if (ov > best[j] || (ov == best[j] && oi < bidx[j])) {
        best[j] = ov; bidx[j] = oi;
      }
    }
  }
  if (r == 0) {                // lanes 0 and 16: rows j and 8+j respectively
    #pragma unroll
    for (int j = 0; j < 8; ++j) {
      const int n = n0 + h * 8 + j;
      idx_i[n] = bidx[j];
      idx_f[n] = (float)bidx[j];
    }
  }
}

// ---------------------------------------------------------------------------
// Kernel 3: gather codebook rows -> out [B, D, T] and per-block SSE partials.
// Block handles 16 rows; thread (c = tid/16, r = tid%16) covers dims c*8..c*8+7
// of row r, so loads/stores along T are coalesced across consecutive tids.
// ---------------------------------------------------------------------------
__global__ __launch_bounds__(256) void vq_quantize(
    const float* __restrict__ x,        // [B, D, T]
    const float* __restrict__ cb,       // raw codebook
    const int*   __restrict__ idx,
    float* __restrict__ out,            // [B, D, T]
    float* __restrict__ partial) {
  const int tid = threadIdx.x;
  const int r = tid & 15;
  const int c = tid >> 4;
  const int n0 = blockIdx.x * 16;
  const int b = n0 >> 11;
  const int t = (n0 & (T_LEN - 1)) + r;
  const int k = idx[n0 + r];
  const float* cbr = cb + (size_t)k * DIM + c * 8;
  const size_t base = ((size_t)b * DIM + c * 8) * T_LEN + t;
  float ssum = 0.0f;
  #pragma unroll
  for (int i = 0; i < 8; ++i) {
    float q  = cbr[i];
    float xv = x[base + (size_t)i * T_LEN];
    out[base + (size_t)i * T_LEN] = q;
    float d = q - xv;
    ssum += d * d;
  }
  __shared__ float red[256];
  red[tid] = ssum;
  __syncthreads();
  for (int s = 128; s > 0; s >>= 1) {
    if (tid < s) red[tid] += red[tid + s];
    __syncthreads();
  }
  if (tid == 0) partial[blockIdx.x] = red[0];
}

// ---------------------------------------------------------------------------
// Kernel 4: reduce 2048 partials -> loss = (1 + 0.02) * mse
// (numerically q_latent_loss == e_latent_loss == mse).
// ---------------------------------------------------------------------------
__global__ __launch_bounds__(256) void vq_loss(
    const float* __restrict__ partial, float* __restrict__ loss) {
  const int tid = threadIdx.x;
  float s = 0.0f;
  #pragma unroll
  for (int i = 0; i < 8; ++i) s += partial[tid + 256 * i];
  __shared__ float red[256];
  red[tid] = s;
  __syncthreads();
  for (int sh = 128; sh > 0; sh >>= 1) {
    if (tid < sh) red[tid] += red[tid + sh];
    __syncthreads();
  }
  if (tid == 0)
    loss[0] = 1.02f * red[0] / (float)((size_t)N_ROWS * DIM);
}

// ---------------------------------------------------------------------------
extern "C" void kernel_launch(void* const* d_in, const int* in_sizes, int n_in,
                              void* d_out, int out_size, void* d_ws, size_t ws_size,
                              hipStream_t stream) {
  const float* x  = (const float*)d_in[0];   // [16,128,2048] f32
  const float* cb = (const float*)d_in[1];   // [8192,128] f32

  float* out_base = (float*)d_out;
  float* loss   = out_base;                       // [1]
  float* quant  = out_base + 1;                   // [B*D*T] = 4194304
  float* idx_f  = quant + (size_t)N_ROWS * DIM;   // [N] indices as float

  float* cbn     = (float*)d_ws;                          // 8192*128 floats
  float* partial = cbn + (size_t)NUM_CODES * DIM;         // 2048 floats
  int*   idx_i   = (int*)(partial + 2048);                // 32768 ints

  vq_normalize_cb<<<NUM_CODES / 8, 256, 0, stream>>>(cb, cbn);
  vq_argmax<<<N_ROWS / 128, 256, 0, stream>>>(x, cbn, idx_i, idx_f);
  vq_quantize<<<N_ROWS / 16, 256, 0, stream>>>(x, cb, idx_i, quant, partial);
  vq_loss<<<1, 256, 0, stream>>>(partial, loss);
}